// GraphAttentionNetwork_25391846654492
// MI455X (gfx1250) — compile-verified
//
#include <hip/hip_runtime.h>
#include <math.h>

// Problem constants (match reference)
#define N_NODES 50000
#define E_EDGES 800000
#define ET_EDGES (E_EDGES + N_NODES)   // with self-loops
#define IN_CH 128
#define OUT_CH 32
#define HEADS 4
#define HO (HEADS * OUT_CH)            // 128
#define NEG_SLOPE 0.2f

typedef float v2f __attribute__((ext_vector_type(2)));
typedef float v8f __attribute__((ext_vector_type(8)));

// ---------------------------------------------------------------------------
// init: out[n][c] = bias[c]; amax = -inf; denom = 0
// ---------------------------------------------------------------------------
__global__ __launch_bounds__(256) void gat_init(const float* __restrict__ bias,
                                                float* __restrict__ out,
                                                float* __restrict__ amax,
                                                float* __restrict__ denom) {
  int idx = blockIdx.x * blockDim.x + threadIdx.x;
  if (idx < N_NODES * HO) {
    out[idx] = bias[idx & (HO - 1)];
  }
  if (idx < N_NODES * HEADS) {
    amax[idx] = -__builtin_huge_valf();
    denom[idx] = 0.0f;
  }
}

// ---------------------------------------------------------------------------
// h = x @ W^T via V_WMMA_F32_16X16X4_F32 (fp32 WMMA, 16x16x4)
//   A[m][k] = x[row0+m][k]            (16x4 frag: lane=(k>=2)*16+m, vgpr=k&1)
//   B[k][n] = W[col0+n][k]            (4x16 frag, mirrored layout)
//   C 16x16: vgpr i -> rows i (lanes 0-15) / i+8 (lanes 16-31), col = lane&15
// Block: 256 threads = 8 waves; wave w owns cols 16w..16w+15 of a 16-row tile.
// 50000 rows = 3125 tiles exactly -> no tails, EXEC all-ones (WMMA req).
// ---------------------------------------------------------------------------
__global__ __launch_bounds__(256) void gat_gemm_wmma(const float* __restrict__ x,
                                                     const float* __restrict__ W,
                                                     float* __restrict__ h) {
  const int tile = blockIdx.x;
  const int wave = threadIdx.x >> 5;      // 0..7
  const int lane = threadIdx.x & 31;

  const int row0 = tile * 16;
  const int col0 = wave * 16;

  const int mr = lane & 15;               // A: row-in-tile, B: col-in-tile
  const int kh = (lane >> 4) * 2;         // K sub-offset 0 or 2

  const float* __restrict__ xrow = x + (size_t)(row0 + mr) * IN_CH;
  const float* __restrict__ wrow = W + (size_t)(col0 + mr) * IN_CH; // B[k][n]=W[n][k]

  v8f acc = {};
#pragma unroll
  for (int k0 = 0; k0 < IN_CH; k0 += 4) {
    v2f a = *(const v2f*)(xrow + k0 + kh);   // 8B aligned: rows are 512B aligned, k even
    v2f b = *(const v2f*)(wrow + k0 + kh);
    acc = __builtin_amdgcn_wmma_f32_16x16x4_f32(
        /*neg_a=*/false, a, /*neg_b=*/false, b,
        /*c_mod=*/(short)0, acc, /*reuse_a=*/false, /*reuse_b=*/false);
  }

  const int rbase = (lane >> 4) * 8;
  const int col = lane & 15;
#pragma unroll
  for (int i = 0; i < 8; ++i) {
    h[(size_t)(row0 + rbase + i) * HO + col0 + col] = acc[i];
  }
}

// ---------------------------------------------------------------------------
// a_src[n][h] = dot(h[n, h*32 : h*32+32], att_src[h]);  same for a_dst
// ---------------------------------------------------------------------------
__global__ __launch_bounds__(256) void gat_attn(const float* __restrict__ h,
                                                const float* __restrict__ att_src,
                                                const float* __restrict__ att_dst,
                                                float* __restrict__ a_src,
                                                float* __restrict__ a_dst) {
  int idx = blockIdx.x * blockDim.x + threadIdx.x;
  if (idx >= N_NODES * HEADS) return;
  int node = idx >> 2;                    // / HEADS
  int hd = idx & (HEADS - 1);
  const float* hp = h + (size_t)node * HO + hd * OUT_CH;
  const float* as = att_src + hd * OUT_CH;
  const float* ad = att_dst + hd * OUT_CH;
  float s = 0.0f, d = 0.0f;
#pragma unroll
  for (int c = 0; c < OUT_CH; ++c) {
    float v = hp[c];
    s = fmaf(v, as[c], s);
    d = fmaf(v, ad[c], d);
  }
  a_src[idx] = s;
  a_dst[idx] = d;
}

// ---------------------------------------------------------------------------
// Edge helpers
// ---------------------------------------------------------------------------
__device__ __forceinline__ void edge_nodes(int e, const int* __restrict__ ei,
                                           int& s, int& d) {
  if (e < E_EDGES) {
    s = ei[e];
    d = ei[E_EDGES + e];
  } else {                                // self-loops appended
    s = d = e - E_EDGES;
  }
}

__device__ __forceinline__ float leaky(float v) {
  return v > 0.0f ? v : v * NEG_SLOPE;
}

// float atomic max via sign-split int/uint trick (IEEE ordering):
//   non-negative values: int compare == float compare
//   negative values: unsigned-min on raw bits == float-max
__device__ __forceinline__ void atomicMaxF32(float* addr, float val) {
  if (val >= 0.0f) {
    atomicMax((int*)addr, __float_as_int(val));
  } else {
    atomicMin((unsigned int*)addr, __float_as_uint(val));
  }
}

// ---------------------------------------------------------------------------
// segment max of leaky(alpha) over incoming edges
// ---------------------------------------------------------------------------
__global__ __launch_bounds__(256) void gat_edge_max(const int* __restrict__ ei,
                                                    const float* __restrict__ a_src,
                                                    const float* __restrict__ a_dst,
                                                    float* __restrict__ amax) {
  int e = blockIdx.x * blockDim.x + threadIdx.x;
  if (e >= ET_EDGES) return;
  int s, d;
  edge_nodes(e, ei, s, d);
#pragma unroll
  for (int hh = 0; hh < HEADS; ++hh) {
    float al = leaky(a_src[s * HEADS + hh] + a_dst[d * HEADS + hh]);
    atomicMaxF32(&amax[d * HEADS + hh], al);
  }
}

// ---------------------------------------------------------------------------
// segment sum of exp(alpha - amax[dst])
// ---------------------------------------------------------------------------
__global__ __launch_bounds__(256) void gat_edge_sum(const int* __restrict__ ei,
                                                    const float* __restrict__ a_src,
                                                    const float* __restrict__ a_dst,
                                                    const float* __restrict__ amax,
                                                    float* __restrict__ denom) {
  int e = blockIdx.x * blockDim.x + threadIdx.x;
  if (e >= ET_EDGES) return;
  int s, d;
  edge_nodes(e, ei, s, d);
#pragma unroll
  for (int hh = 0; hh < HEADS; ++hh) {
    float al = leaky(a_src[s * HEADS + hh] + a_dst[d * HEADS + hh]);
    float ex = expf(al - amax[d * HEADS + hh]);
    atomicAdd(&denom[d * HEADS + hh], ex);
  }
}

// ---------------------------------------------------------------------------
// scatter: one wave32 per edge; lane owns 4 channels (float4 of h[src]),
// weight w depends only on lane's head (= lane/8). 4x global_atomic_add_f32.
// ---------------------------------------------------------------------------
__global__ __launch_bounds__(256) void gat_scatter(const float* __restrict__ h,
                                                   const int* __restrict__ ei,
                                                   const float* __restrict__ a_src,
                                                   const float* __restrict__ a_dst,
                                                   const float* __restrict__ amax,
                                                   const float* __restrict__ denom,
                                                   float* __restrict__ out) {
  int wid = blockIdx.x * (blockDim.x >> 5) + (threadIdx.x >> 5);
  if (wid >= ET_EDGES) return;
  int lane = threadIdx.x & 31;
  int s, d;
  edge_nodes(wid, ei, s, d);

  int hh = lane >> 3;                     // channels 4*lane..4*lane+3 -> head lane/8
  float al = leaky(a_src[s * HEADS + hh] + a_dst[d * HEADS + hh]);
  float w = expf(al - amax[d * HEADS + hh]) / denom[d * HEADS + hh];

  const float4 hv = *(const float4*)(h + (size_t)s * HO + lane * 4);
  float* op = out + (size_t)d * HO + lane * 4;
  atomicAdd(op + 0, hv.x * w);
  atomicAdd(op + 1, hv.y * w);
  atomicAdd(op + 2, hv.z * w);
  atomicAdd(op + 3, hv.w * w);
}

// ---------------------------------------------------------------------------
// launch
// ---------------------------------------------------------------------------
extern "C" void kernel_launch(void* const* d_in, const int* in_sizes, int n_in,
                              void* d_out, int out_size, void* d_ws, size_t ws_size,
                              hipStream_t stream) {
  const float* x       = (const float*)d_in[0];
  const int*   ei      = (const int*)d_in[1];
  const float* W       = (const float*)d_in[2];
  const float* att_src = (const float*)d_in[3];
  const float* att_dst = (const float*)d_in[4];
  const float* bias    = (const float*)d_in[5];
  float* out = (float*)d_out;

  // workspace layout (floats): h[N*128] | a_src[N*4] | a_dst[N*4] | amax[N*4] | denom[N*4]
  float* h      = (float*)d_ws;
  float* a_src  = h + (size_t)N_NODES * HO;
  float* a_dst  = a_src + (size_t)N_NODES * HEADS;
  float* amax   = a_dst + (size_t)N_NODES * HEADS;
  float* denom  = amax + (size_t)N_NODES * HEADS;

  (void)in_sizes; (void)n_in; (void)out_size; (void)ws_size;

  // 1) init out/bias, amax, denom
  {
    int total = N_NODES * HO;
    gat_init<<<(total + 255) / 256, 256, 0, stream>>>(bias, out, amax, denom);
  }
  // 2) h = x @ W^T  (fp32 WMMA)
  gat_gemm_wmma<<<N_NODES / 16, 256, 0, stream>>>(x, W, h);
  // 3) attention dots
  {
    int total = N_NODES * HEADS;
    gat_attn<<<(total + 255) / 256, 256, 0, stream>>>(h, att_src, att_dst, a_src, a_dst);
  }
  // 4) segment max
  gat_edge_max<<<(ET_EDGES + 255) / 256, 256, 0, stream>>>(ei, a_src, a_dst, amax);
  // 5) segment sum
  gat_edge_sum<<<(ET_EDGES + 255) / 256, 256, 0, stream>>>(ei, a_src, a_dst, amax, denom);
  // 6) weighted scatter (wave per edge, 8 waves per block)
  gat_scatter<<<(ET_EDGES + 7) / 8, 256, 0, stream>>>(h, ei, a_src, a_dst, amax, denom, out);
}